// MatchMatrix_23605140259564
// MI455X (gfx1250) — compile-verified
//
#include <hip/hip_runtime.h>

typedef __attribute__((ext_vector_type(2))) float v2f;
typedef __attribute__((ext_vector_type(8))) float v8f;

__device__ __forceinline__ v8f wmma4(v2f a, v2f b, v8f c) {
  // V_WMMA_F32_16X16X4_F32: A 16x4, B 4x16, C/D 16x16 f32
  return __builtin_amdgcn_wmma_f32_16x16x4_f32(false, a, false, b, (short)0, c,
                                               false, false);
}

__device__ __forceinline__ float leaky(float x) {
  return x >= 0.0f ? x : 0.01f * x;
}

// ---------------------------------------------------------------------------
// Kernel 1: a_pb  = sent_a(1024,256) @ W1[0:256]  + b1   -> (1024,64)
//           b_part= sent_b(1024,256) @ W1[256:512]       -> (1024,64)
// One wave per 16x16 output tile; K-loop over 256 in steps of 4.
// ---------------------------------------------------------------------------
__global__ void __launch_bounds__(128) proj_kernel(
    const float* __restrict__ sa, const float* __restrict__ sb,
    const float* __restrict__ W1, const float* __restrict__ b1,
    float* __restrict__ a_pb, float* __restrict__ b_part) {
  const int lane = threadIdx.x & 31;
  const int lo = lane & 15, hi = lane >> 4;
  const int wave = (blockIdx.x * 128 + threadIdx.x) >> 5;  // 0..511
  const int mat = wave >> 8;                               // 0=a, 1=b
  const int t = wave & 255;
  const int m0 = (t >> 2) * 16;
  const int n0 = (t & 3) * 16;

  const float* src = mat ? sb : sa;                 // (1024, 256)
  const float* W   = W1 + (mat ? 256 * 64 : 0);     // (256, 64) slice
  const float* arow = src + (size_t)(m0 + lo) * 256 + 2 * hi;
  const float* wcol = W + n0 + lo + (size_t)(2 * hi) * 64;

  v8f c = {};
#pragma unroll 8
  for (int kk = 0; kk < 256; kk += 4) {
    v2f a = *(const v2f*)(arow + kk);               // A[lo][kk+2hi .. +1]
    v2f b;
    b.x = wcol[(size_t)kk * 64];                    // B[kk+2hi][n0+lo]
    b.y = wcol[(size_t)(kk + 1) * 64];              // B[kk+2hi+1][n0+lo]
    c = wmma4(a, b, c);
  }

  const float bias = mat ? 0.0f : b1[n0 + lo];      // fold b1 into a-part
  float* dst = (mat ? b_part : a_pb) + n0 + lo;
#pragma unroll
  for (int r = 0; r < 8; ++r)
    dst[(size_t)(m0 + r + 8 * hi) * 64] = c[r] + bias;
}

// ---------------------------------------------------------------------------
// Kernel 2: fused  h = leaky(a_pb[b,a,:] + b_part[b,c,:])   (16x64 tile)
//                  h2 = leaky(h @ W2 + b2)                  (16x32, via LDS)
//                  out= leaky(h2 @ W3 + b3)                 (16x16)
// One wave per (b, a, c-half); 4 c-tiles of 16 per wave.
// ---------------------------------------------------------------------------
#define LSTR 34  // padded LDS row stride (floats), avoids bank conflicts

__global__ void __launch_bounds__(128) match_mlp_kernel(
    const float* __restrict__ a_pb, const float* __restrict__ b_part,
    const float* __restrict__ W2, const float* __restrict__ b2,
    const float* __restrict__ W3, const float* __restrict__ b3,
    float* __restrict__ out) {
  __shared__ float lds[4 * 16 * LSTR];
  const int lane = threadIdx.x & 31;
  const int lo = lane & 15, hi = lane >> 4;
  const int wave = (blockIdx.x * 128 + threadIdx.x) >> 5;  // 0..2047
  const int ba = wave >> 1;        // b*128 + a   (0..1023)
  const int half = wave & 1;       // which 64 columns of c
  const int bbase = ba & ~127;     // b*128
  float* wlds = lds + (threadIdx.x >> 5) * (16 * LSTR);

  // --- per-wave preloads (amortized over 4 c-tiles) ---
  v2f abv[16];                                     // a_pb row (b1 folded in)
  const float* apr = a_pb + (size_t)ba * 64 + 2 * hi;
#pragma unroll
  for (int i = 0; i < 16; ++i) abv[i] = *(const v2f*)(apr + 4 * i);

  v2f w2f[2][16];                                  // W2 B-fragments (64x32)
#pragma unroll
  for (int t = 0; t < 2; ++t) {
    const float* w = W2 + 16 * t + lo + (size_t)(2 * hi) * 32;
#pragma unroll
    for (int i = 0; i < 16; ++i) {
      w2f[t][i].x = w[(size_t)(4 * i) * 32];
      w2f[t][i].y = w[(size_t)(4 * i + 1) * 32];
    }
  }
  v2f w3f[8];                                      // W3 B-fragments (32x16)
  {
    const float* w = W3 + lo + (size_t)(2 * hi) * 16;
#pragma unroll
    for (int i = 0; i < 8; ++i) {
      w3f[i].x = w[(size_t)(4 * i) * 16];
      w3f[i].y = w[(size_t)(4 * i + 1) * 16];
    }
  }
  const float b2v0 = b2[lo], b2v1 = b2[16 + lo];
  const float b3v = b3[lo];

  for (int it = 0; it < 4; ++it) {
    const int c0 = (half * 4 + it) * 16;

    // h fragments: A-operand layout, lane lo = row c, hi selects K pair
    const float* bpr = b_part + (size_t)(bbase + c0 + lo) * 64 + 2 * hi;
    v2f hf[16];
#pragma unroll
    for (int i = 0; i < 16; ++i) {
      v2f bp = *(const v2f*)(bpr + 4 * i);
      hf[i].x = leaky(abv[i].x + bp.x);
      hf[i].y = leaky(abv[i].y + bp.y);
    }

    // layer 2: (16x64) @ (64x32) -> two 16x16 accumulators
    v8f c2a = {}, c2b = {};
#pragma unroll
    for (int i = 0; i < 16; ++i) {
      c2a = wmma4(hf[i], w2f[0][i], c2a);
      c2b = wmma4(hf[i], w2f[1][i], c2b);
    }

    // bias + leaky, park into LDS row-major (c, n) for the transpose
#pragma unroll
    for (int r = 0; r < 8; ++r) {
      wlds[(r + 8 * hi) * LSTR + lo]      = leaky(c2a[r] + b2v0);
      wlds[(r + 8 * hi) * LSTR + 16 + lo] = leaky(c2b[r] + b2v1);
    }
    asm volatile("s_wait_dscnt 0x0" ::: "memory");

    // reload as A fragments for layer 3 (D-layout -> A-layout transpose)
    v2f hf3[8];
#pragma unroll
    for (int i = 0; i < 8; ++i)
      hf3[i] = *(const v2f*)(wlds + lo * LSTR + 4 * i + 2 * hi);
    asm volatile("s_wait_dscnt 0x0" ::: "memory");

    // layer 3: (16x32) @ (32x16)
    v8f c3 = {};
#pragma unroll
    for (int i = 0; i < 8; ++i) c3 = wmma4(hf3[i], w3f[i], c3);

    // out[(b*128+a)*128 + c][n], lanes 0..15 = consecutive n (coalesced)
    float* op = out + ((size_t)ba * 128 + c0) * 16 + lo;
#pragma unroll
    for (int r = 0; r < 8; ++r)
      op[(size_t)(r + 8 * hi) * 16] = leaky(c3[r] + b3v);
  }
}

extern "C" void kernel_launch(void* const* d_in, const int* in_sizes, int n_in,
                              void* d_out, int out_size, void* d_ws,
                              size_t ws_size, hipStream_t stream) {
  (void)in_sizes; (void)n_in; (void)out_size; (void)ws_size;
  const float* sa = (const float*)d_in[0];   // (8,128,256)
  const float* sb = (const float*)d_in[1];   // (8,128,256)
  const float* W1 = (const float*)d_in[2];   // (512,64)
  const float* b1 = (const float*)d_in[3];   // (64,)
  const float* W2 = (const float*)d_in[4];   // (64,32)
  const float* b2 = (const float*)d_in[5];   // (32,)
  const float* W3 = (const float*)d_in[6];   // (32,16)
  const float* b3 = (const float*)d_in[7];   // (16,)
  float* out = (float*)d_out;                // (8,128,128,16)

  float* a_pb   = (float*)d_ws;              // (1024,64) = 256 KB
  float* b_part = a_pb + 1024 * 64;          // (1024,64) = 256 KB

  // 512 waves (one per 16x16 tile of the two projections)
  proj_kernel<<<128, 128, 0, stream>>>(sa, sb, W1, b1, a_pb, b_part);
  // 2048 waves (one per (b, a, c-half)), fused MLP
  match_mlp_kernel<<<512, 128, 0, stream>>>(a_pb, b_part, W2, b2, W3, b3, out);
}